// RelativeStrengthIndex_37658273251538
// MI455X (gfx1250) — compile-verified
//
#include <hip/hip_runtime.h>
#include <hip/hip_bf16.h>

// ---- problem geometry (from setup_inputs: (2048, 8192) f32, window_size=14) ----
#define ROWS     2048
#define N        8192
#define PLEN     (N - 1)              // 8191 percent-change values per row
#define WIN      13                   // window_size - 1
#define OUTLEN   (PLEN - WIN + 1)     // 8179 outputs per row
#define CH       2048                 // outputs per workgroup chunk
#define NCHUNK   4                    // 4 * 2048 >= 8179
#define CHIN     (CH + WIN + 1)       // 2062 inputs needed per chunk (incl. halo)
#define CHIN_PAD 2064                 // padded to multiple of 4 floats (b128 loads)
#define PMAX     (CH + WIN - 1)       // 2060 p-values needed per chunk
#define PMAX_PAD 2064
#define TPB      256                  // 8 wave32 per block

__global__ __launch_bounds__(TPB)
void rsi_kernel(const float* __restrict__ in, float* __restrict__ out) {
    __shared__ float s_in[CHIN_PAD];   // staged inputs (written by async DMA)
    __shared__ float s_g[PMAX_PAD];    // gains
    __shared__ float s_l[PMAX_PAD];    // losses
    __shared__ float s_out[CH];        // restage for coalesced global stores

    const int tid  = threadIdx.x;
    const int bid  = blockIdx.x;
    const int row  = bid >> 2;
    const int c    = bid & 3;
    const int cb   = c * CH;

    const int in_count  = (N - cb)      < CHIN ? (N - cb)      : CHIN;   // 2062 or 2048
    const int p_count   = (PLEN - cb)   < PMAX ? (PLEN - cb)   : PMAX;   // 2060 or 2047
    const int out_count = (OUTLEN - cb) < CH   ? (OUTLEN - cb) : CH;     // 2048 or 2035

    // -------- Stage 1: async global -> LDS staging (CDNA5 async data mover) ----
    {
        // low 32 bits of a generic shared pointer == LDS byte address
        const unsigned lds_base = (unsigned)(uintptr_t)(&s_in[0]);
        const float* gbase = in + (size_t)row * N + cb;
        const int nvec = (in_count + 3) >> 2;          // 516 or 512 float4s, 16B aligned
        for (int v = tid; v < nvec; v += TPB) {
            unsigned long long ga = (unsigned long long)(uintptr_t)(gbase + 4 * v);
            unsigned           la = lds_base + 16u * (unsigned)v;
            asm volatile("global_load_async_to_lds_b128 %0, %1, off"
                         :: "v"(la), "v"(ga) : "memory");
        }
        asm volatile("s_wait_asynccnt 0" ::: "memory");
    }
    __syncthreads();

    // -------- Stage 2: percent change -> gains / losses (float4 granularity) ---
    {
        const int ngrp = (p_count + 3) >> 2;
        for (int g = tid; g < ngrp; g += TPB) {
            const int i4 = g << 2;
            float4 x = *reinterpret_cast<const float4*>(&s_in[i4]);
            float  x4 = s_in[i4 + 4];
            float xv[5] = { x.x, x.y, x.z, x.w, x4 };
            float gg[4], ll[4];
#pragma unroll
            for (int e = 0; e < 4; ++e) {
                float prev = xv[e];
                float nxt  = xv[e + 1];
                float pr   = (prev != 0.0f) ? (nxt - prev) / prev : 0.0f;
                gg[e] = pr > 0.0f ?  pr : 0.0f;
                ll[e] = pr < 0.0f ? -pr : 0.0f;
            }
            if (i4 + 4 <= p_count) {
                *reinterpret_cast<float4*>(&s_g[i4]) = make_float4(gg[0], gg[1], gg[2], gg[3]);
                *reinterpret_cast<float4*>(&s_l[i4]) = make_float4(ll[0], ll[1], ll[2], ll[3]);
            } else {
#pragma unroll
                for (int e = 0; e < 4; ++e)
                    if (i4 + e < p_count) { s_g[i4 + e] = gg[e]; s_l[i4 + e] = ll[e]; }
            }
        }
        // zero the tail so sliding windows past p_count see defined values
        for (int i = p_count + tid; i < PMAX_PAD; i += TPB) {
            s_g[i] = 0.0f;
            s_l[i] = 0.0f;
        }
    }
    __syncthreads();

    // -------- Stage 3: sliding window-13 sums, 8 contiguous outputs / thread ---
    {
        const int j0 = tid << 3;
        if (j0 < out_count) {
            float a[20], b[20];
#pragma unroll
            for (int q = 0; q < 5; ++q) {              // 16B-aligned ds b128 reads
                float4 vg = *reinterpret_cast<const float4*>(&s_g[j0 + 4 * q]);
                float4 vl = *reinterpret_cast<const float4*>(&s_l[j0 + 4 * q]);
                a[4*q+0] = vg.x; a[4*q+1] = vg.y; a[4*q+2] = vg.z; a[4*q+3] = vg.w;
                b[4*q+0] = vl.x; b[4*q+1] = vl.y; b[4*q+2] = vl.z; b[4*q+3] = vl.w;
            }
            float sg = 0.0f, sl = 0.0f;
#pragma unroll
            for (int i = 0; i < WIN; ++i) { sg += a[i]; sl += b[i]; }
#pragma unroll
            for (int k = 0; k < 8; ++k) {
                if (k) {                                // slide: +newest -oldest
                    sg += a[WIN + k - 1] - a[k - 1];
                    sl += b[WIN + k - 1] - b[k - 1];
                }
                if (j0 + k < out_count) {
                    // avg_gain/avg_loss: the 1/13 factors cancel in the ratio
                    float rs = (sl != 0.0f) ? (sg / sl) : 0.0f;
                    s_out[j0 + k] = 1.0f - 1.0f / (1.0f + rs);
                }
            }
        }
    }
    __syncthreads();

    // -------- Stage 4: coalesced global stores (stride-1 across lanes) --------
    {
        float* obase = out + (size_t)row * OUTLEN + cb;
        for (int i = tid; i < out_count; i += TPB) obase[i] = s_out[i];
    }
}

extern "C" void kernel_launch(void* const* d_in, const int* in_sizes, int n_in,
                              void* d_out, int out_size, void* d_ws, size_t ws_size,
                              hipStream_t stream) {
    (void)in_sizes; (void)n_in; (void)out_size; (void)d_ws; (void)ws_size;
    const float* in  = (const float*)d_in[0];   // (2048, 8192) f32
    float*       out = (float*)d_out;           // (2048, 8179) f32
    rsi_kernel<<<dim3(ROWS * NCHUNK), dim3(TPB), 0, stream>>>(in, out);
}